// ChamferDistance_86002425135442
// MI455X (gfx1250) — compile-verified
//
#include <hip/hip_runtime.h>

// CDNA5 / gfx1250: wave32, WMMA fp32 16x16x4.
typedef __attribute__((ext_vector_type(2))) float v2f;
typedef __attribute__((ext_vector_type(8))) float v8f;

#define N_PTS 16384
#define K_DIM 64
#define TILE  128
#define INF_BITS 0x7F800000u

// ---------------------------------------------------------------------------
// Kernel 1: per-row squared norms for a and b + init global min arrays to +inf
// tid in [0, 2N): first half handles a/rowmin, second half b/colmin.
// ---------------------------------------------------------------------------
__global__ void cd_norms_init(const float* __restrict__ a,
                              const float* __restrict__ b,
                              unsigned* __restrict__ rowmin,
                              unsigned* __restrict__ colmin,
                              float* __restrict__ anorm,
                              float* __restrict__ bnorm) {
  int tid = blockIdx.x * blockDim.x + threadIdx.x;
  const float* src;
  float* dst;
  unsigned* mn;
  int idx;
  if (tid < N_PTS) {
    src = a; dst = anorm; mn = rowmin; idx = tid;
  } else {
    src = b; dst = bnorm; mn = colmin; idx = tid - N_PTS;
  }
  const float4* row = reinterpret_cast<const float4*>(src + (size_t)idx * K_DIM);
  float s = 0.0f;
#pragma unroll
  for (int i = 0; i < K_DIM / 4; ++i) {
    float4 v = row[i];
    s = fmaf(v.x, v.x, s);
    s = fmaf(v.y, v.y, s);
    s = fmaf(v.z, v.z, s);
    s = fmaf(v.w, v.w, s);
  }
  dst[idx] = s;
  mn[idx] = INF_BITS;
}

// ---------------------------------------------------------------------------
// Kernel 2: fused tiled distance^2 + row/col min.
// Workgroup = 256 threads = 8 waves. Wave grid 4x2 (M x N); each wave computes
// a 32x64 output tile as 2x4 WMMA 16x16 accumulators over K=64 (16 steps of
// K=4 with V_WMMA_F32_16X16X4_F32). Mins reduced through LDS, then one
// global atomicMin per output row/col per workgroup.
//
// f32 WMMA fragment layout (ISA 7.12.2):
//   A 16x4 : lane (half h, r=lane&15): v0 = A[r][k+2h], v1 = A[r][k+2h+1]
//   B 4x16 : mirrored with N on lanes:  v0 = B[k+2h][r], v1 = B[k+2h+1][r]
//   C/D    : VGPR t holds M = t + 8h, lane r holds N = r
// Both A and B fragments are 8-byte contiguous row-major loads from a and b.
// ---------------------------------------------------------------------------
__global__ __launch_bounds__(256)
void cd_tile(const float* __restrict__ a, const float* __restrict__ b,
             const float* __restrict__ anorm, const float* __restrict__ bnorm,
             unsigned* __restrict__ rowmin, unsigned* __restrict__ colmin) {
  __shared__ unsigned sRow[TILE];
  __shared__ unsigned sCol[TILE];

  const int tid = threadIdx.x;
  if (tid < TILE) {
    sRow[tid] = INF_BITS;
    sCol[tid] = INF_BITS;
  }
  __syncthreads();

  const int wave  = tid >> 5;   // 0..7
  const int lane  = tid & 31;
  const int half  = lane >> 4;  // 0/1
  const int r     = lane & 15;
  const int waveM = wave >> 1;  // 0..3 : 32 rows each
  const int waveN = wave & 1;   // 0..1 : 64 cols each

  const int rowBase = blockIdx.y * TILE + waveM * 32;
  const int colBase = blockIdx.x * TILE + waveN * 64;

  v8f acc[2][4];
  const v8f vzero = {};
#pragma unroll
  for (int mi = 0; mi < 2; ++mi)
#pragma unroll
    for (int ni = 0; ni < 4; ++ni) acc[mi][ni] = vzero;

  const float* aPtr0 = a + (size_t)(rowBase + r) * K_DIM + 2 * half;
  const float* aPtr1 = aPtr0 + 16 * K_DIM;
  const float* bPtr[4];
#pragma unroll
  for (int ni = 0; ni < 4; ++ni)
    bPtr[ni] = b + (size_t)(colBase + ni * 16 + r) * K_DIM + 2 * half;

#pragma unroll
  for (int k = 0; k < K_DIM; k += 4) {
    v2f av[2], bv[4];
    av[0] = *reinterpret_cast<const v2f*>(aPtr0 + k);
    av[1] = *reinterpret_cast<const v2f*>(aPtr1 + k);
#pragma unroll
    for (int ni = 0; ni < 4; ++ni)
      bv[ni] = *reinterpret_cast<const v2f*>(bPtr[ni] + k);

#pragma unroll
    for (int mi = 0; mi < 2; ++mi)
#pragma unroll
      for (int ni = 0; ni < 4; ++ni)
        acc[mi][ni] = __builtin_amdgcn_wmma_f32_16x16x4_f32(
            /*neg_a=*/false, av[mi], /*neg_b=*/false, bv[ni],
            /*c_mod=*/(short)0, acc[mi][ni],
            /*reuse_a=*/false, /*reuse_b=*/false);
  }

  // Epilogue: sq = |a|^2 + |b|^2 - 2 dot  (clamped at 0); min over tile.
  // sqrt deferred to the final reduction (monotonic).
  float bn[4];
#pragma unroll
  for (int ni = 0; ni < 4; ++ni) bn[ni] = bnorm[colBase + ni * 16 + r];

  float cmin[4] = {3.402823466e+38f, 3.402823466e+38f,
                   3.402823466e+38f, 3.402823466e+38f};

#pragma unroll
  for (int mi = 0; mi < 2; ++mi) {
#pragma unroll
    for (int t = 0; t < 8; ++t) {
      const int mOff = mi * 16 + 8 * half + t;          // 0..31 within wave rows
      const float an = anorm[rowBase + mOff];
      float rmin = 3.402823466e+38f;
#pragma unroll
      for (int ni = 0; ni < 4; ++ni) {
        float sq = fmaxf(fmaf(-2.0f, acc[mi][ni][t], an + bn[ni]), 0.0f);
        rmin = fminf(rmin, sq);
        cmin[ni] = fminf(cmin[ni], sq);
      }
      atomicMin(&sRow[waveM * 32 + mOff], __float_as_uint(rmin));
    }
  }
#pragma unroll
  for (int ni = 0; ni < 4; ++ni)
    atomicMin(&sCol[waveN * 64 + ni * 16 + r], __float_as_uint(cmin[ni]));

  __syncthreads();
  if (tid < TILE) {
    atomicMin(&rowmin[blockIdx.y * TILE + tid], sRow[tid]);
    atomicMin(&colmin[blockIdx.x * TILE + tid], sCol[tid]);
  }
}

// ---------------------------------------------------------------------------
// Kernel 3: mean of sqrt over the 2N minima -> scalar output.
// ---------------------------------------------------------------------------
__global__ void cd_reduce(const unsigned* __restrict__ mins,
                          float* __restrict__ out) {
  __shared__ float sdata[256];
  float s = 0.0f;
  for (int i = threadIdx.x; i < 2 * N_PTS; i += 256)
    s += sqrtf(__uint_as_float(mins[i]));
  sdata[threadIdx.x] = s;
  __syncthreads();
  for (int off = 128; off > 0; off >>= 1) {
    if ((int)threadIdx.x < off) sdata[threadIdx.x] += sdata[threadIdx.x + off];
    __syncthreads();
  }
  if (threadIdx.x == 0) out[0] = sdata[0] / (float)(2 * N_PTS);
}

// ---------------------------------------------------------------------------
extern "C" void kernel_launch(void* const* d_in, const int* in_sizes, int n_in,
                              void* d_out, int out_size, void* d_ws,
                              size_t ws_size, hipStream_t stream) {
  const float* a = (const float*)d_in[0];
  const float* b = (const float*)d_in[1];

  // Workspace layout: rowmin[N] | colmin[N] | anorm[N] | bnorm[N]  (256 KB)
  unsigned* rowmin = (unsigned*)d_ws;
  unsigned* colmin = rowmin + N_PTS;
  float* anorm = (float*)(colmin + N_PTS);
  float* bnorm = anorm + N_PTS;

  cd_norms_init<<<dim3((2 * N_PTS) / 256), 256, 0, stream>>>(
      a, b, rowmin, colmin, anorm, bnorm);

  cd_tile<<<dim3(N_PTS / TILE, N_PTS / TILE), 256, 0, stream>>>(
      a, b, anorm, bnorm, rowmin, colmin);

  cd_reduce<<<1, 256, 0, stream>>>(rowmin, (float*)d_out);
}